// Network_78872779423820
// MI455X (gfx1250) — compile-verified
//
#include <hip/hip_runtime.h>
#include <cstdint>

// Native vector type so __builtin_nontemporal_load/store accept it and we get b128 ops.
typedef float v4f __attribute__((ext_vector_type(4)));

#define TPB   256
#define ITEMS 2      // float4s per thread per round (32 B/lane in flight per tile)

// Quantizer constants (Z_MIN=-1, Z_MAX=1, Z_LEVEL=25): step = 2/24.
// Forward value of the straight-through estimator is hard_q = nearest level.
__device__ __forceinline__ float quant1(float v) {
    // k = round_to_nearest_even((x - zmin) / step) = rne(fma(x, 12, 12))
    float t = __builtin_rintf(fmaf(v, 12.0f, 12.0f));     // v_rndne_f32
    t = fminf(fmaxf(t, 0.0f), 24.0f);                     // clamp to [0, L-1]
    // match jax: levels = arange * step + zmin, separate mul/add (no fma contraction)
    const float step = 2.0f / 24.0f;                      // 0x3DAAAAAB
    return __fadd_rn(__fmul_rn(t, step), -1.0f);
}

__device__ __forceinline__ v4f quant4(v4f v) {
    v4f o;
    o.x = quant1(v.x); o.y = quant1(v.y); o.z = quant1(v.z); o.w = quant1(v.w);
    return o;
}

// One 16-byte async global->LDS copy for this lane (ASYNCcnt path, gfx1250 TDM-side).
#define ASYNC_LOAD_B128(lds_u32, gaddr_u64)                                   \
    asm volatile("global_load_async_to_lds_b128 %0, %1, off"                  \
                 :: "v"(lds_u32), "v"(gaddr_u64) : "memory")

__global__ __launch_bounds__(TPB) void quantize_st_kernel(
    const v4f* __restrict__ x4, v4f* __restrict__ out4, unsigned n4,
    const float* __restrict__ x, float* __restrict__ out, unsigned n,
    unsigned rounds)                                  // n4 / (gridDim*TPB*ITEMS), host-computed
{
    __shared__ v4f buf[2][ITEMS][TPB];                // double buffer: 16 KB LDS

    const unsigned tid       = threadIdx.x;
    const unsigned nthreads  = gridDim.x * TPB;
    const unsigned per_round = nthreads * ITEMS;      // float4s consumed per round
    const unsigned base      = blockIdx.x * (TPB * ITEMS) + tid;

    // Generic address of __shared__: low 32 bits are the LDS byte address.
    unsigned lds_addr[2][ITEMS];
#pragma unroll
    for (unsigned b = 0; b < 2; ++b)
#pragma unroll
        for (unsigned it = 0; it < ITEMS; ++it)
            lds_addr[b][it] = (unsigned)(uintptr_t)(void*)&buf[b][it][tid];

    // ---- double-buffered async pipeline over full tiles ----
    if (rounds > 0u) {
#pragma unroll
        for (unsigned it = 0; it < ITEMS; ++it) {
            unsigned long long g =
                (unsigned long long)(uintptr_t)(x4 + base + it * TPB);
            ASYNC_LOAD_B128(lds_addr[0][it], g);
        }
    }
    unsigned i = base;
    for (unsigned r = 0; r < rounds; ++r) {           // rounds is grid-uniform
        if (r + 1u < rounds) {
            const unsigned nb = (r + 1u) & 1u;
#pragma unroll
            for (unsigned it = 0; it < ITEMS; ++it) {
                unsigned long long g =
                    (unsigned long long)(uintptr_t)(x4 + i + per_round + it * TPB);
                ASYNC_LOAD_B128(lds_addr[nb][it], g);
            }
            // 4 outstanding; <=2 left means tile r's pair completed (in-order).
            asm volatile("s_wait_asynccnt 0x2" ::: "memory");
        } else {
            asm volatile("s_wait_asynccnt 0x0" ::: "memory");
        }
        const unsigned cb = r & 1u;
        v4f v0 = buf[cb][0][tid];                     // ds_load_b128
        v4f v1 = buf[cb][1][tid];
        __builtin_nontemporal_store(quant4(v0), out4 + i);
        __builtin_nontemporal_store(quant4(v1), out4 + i + TPB);
        i += per_round;
    }

    // ---- float4 tail (n4 not a multiple of per_round): plain grid-stride ----
    for (unsigned j = rounds * per_round + blockIdx.x * TPB + tid; j < n4;
         j += nthreads) {
        v4f v = __builtin_nontemporal_load(x4 + j);
        __builtin_nontemporal_store(quant4(v), out4 + j);
    }

    // ---- scalar tail (n not a multiple of 4): block 0 only ----
    unsigned sidx = n4 * 4u + blockIdx.x * TPB + tid;
    if (sidx < n) {
        out[sidx] = quant1(x[sidx]);
    }
}

extern "C" void kernel_launch(void* const* d_in, const int* in_sizes, int n_in,
                              void* d_out, int out_size, void* d_ws, size_t ws_size,
                              hipStream_t stream) {
    const float* x  = (const float*)d_in[0];
    float* outp     = (float*)d_out;
    const unsigned n  = (unsigned)in_sizes[0];        // 4*64*256*256 = 16,777,216
    const unsigned n4 = n / 4u;                       // 4,194,304 float4s (2^22)

    // per_round = blocks*TPB*ITEMS; 2048*256*2 = 2^20 divides 2^22 exactly:
    // rounds == 4 per thread, zero tail, all waits uniform.
    unsigned blocks = 2048u;
    while (blocks > 1u && blocks * TPB * ITEMS > n4) blocks >>= 1u;
    const unsigned rounds = n4 / (blocks * TPB * ITEMS);

    quantize_st_kernel<<<dim3(blocks), dim3(TPB), 0, stream>>>(
        (const v4f*)x, (v4f*)outp, n4, x, outp, n, rounds);
}